// GCN_molecule_classification_10230612099848
// MI455X (gfx1250) — compile-verified
//
#include <hip/hip_runtime.h>
#include <hip/hip_bf16.h>

typedef __attribute__((ext_vector_type(2))) float v2f;
typedef __attribute__((ext_vector_type(8))) float v8f;

#define N_NODES 100000
#define N_EDGES 1600000
#define HID     64
#define NGRAPH  2048

// ---------------------------------------------------------------- utilities
__global__ void fill_k(float* __restrict__ p, float v, long long n) {
    long long i = (long long)blockIdx.x * blockDim.x + threadIdx.x;
    if (i < n) p[i] = v;
}

__global__ void deg_k(const int* __restrict__ dst, float* __restrict__ deg, int E) {
    int e = blockIdx.x * blockDim.x + threadIdx.x;
    if (e < E) atomicAdd(&deg[dst[e]], 1.0f);
}

__global__ void rsqrt_k(float* __restrict__ d, int n) {
    int i = blockIdx.x * blockDim.x + threadIdx.x;
    if (i < n) d[i] = rsqrtf(d[i]);
}

__global__ void norm_k(const int* __restrict__ src, const int* __restrict__ dst,
                       const float* __restrict__ dis, float* __restrict__ nrm, int E) {
    int e = blockIdx.x * blockDim.x + threadIdx.x;
    if (e < E) nrm[e] = dis[src[e]] * dis[dst[e]];
}

// ---------------------------------------------------------------- WMMA GEMM
// C[rows,64] = A[rows,K] @ W[K,64], fp32 via V_WMMA_F32_16X16X4_F32.
// One wave computes a 16x64 output stripe (4 accumulators) to reuse the A
// fragment 4x. rows is a multiple of 16 (100000 = 6250*16).
__global__ void gemm_wmma_k(const float* __restrict__ A, const float* __restrict__ W,
                            float* __restrict__ C, int rows, int K) {
    const int lane = threadIdx.x & 31;
    const int w    = (blockIdx.x * blockDim.x + threadIdx.x) >> 5;   // wave id
    const int rowTiles = rows >> 4;
    if (w >= rowTiles) return;                                       // wave-uniform exit

    const int m0  = w << 4;
    const int mr  = lane & 15;
    const int khi = (lane >> 4) << 1;                                // 0 or 2

    const float* arow = A + (size_t)(m0 + mr) * K + khi;             // A 16x4 frag base
    const float* bcol = W + mr;                                      // B 4x16 frag base

    v8f acc0 = {}, acc1 = {}, acc2 = {}, acc3 = {};
    for (int k = 0; k < K; k += 4) {
        v2f a;
        const float2 av = *(const float2*)(arow + k);                // 8B aligned
        a[0] = av.x; a[1] = av.y;

        const float* brow0 = bcol + (size_t)(k + khi) * HID;
        const float* brow1 = brow0 + HID;
        v2f b0, b1, b2, b3;
        b0[0] = brow0[0];  b0[1] = brow1[0];
        b1[0] = brow0[16]; b1[1] = brow1[16];
        b2[0] = brow0[32]; b2[1] = brow1[32];
        b3[0] = brow0[48]; b3[1] = brow1[48];

        acc0 = __builtin_amdgcn_wmma_f32_16x16x4_f32(false, a, false, b0, (short)0, acc0, false, false);
        acc1 = __builtin_amdgcn_wmma_f32_16x16x4_f32(false, a, false, b1, (short)0, acc1, false, false);
        acc2 = __builtin_amdgcn_wmma_f32_16x16x4_f32(false, a, false, b2, (short)0, acc2, false, false);
        acc3 = __builtin_amdgcn_wmma_f32_16x16x4_f32(false, a, false, b3, (short)0, acc3, false, false);
    }

    // C/D layout: VGPR j -> row m0+j (lanes 0-15) / m0+8+j (lanes 16-31), col = n0 + (lane&15)
    float* crow = C + (size_t)(m0 + ((lane >> 4) << 3)) * HID + mr;
#pragma unroll
    for (int j = 0; j < 8; ++j) {
        crow[(size_t)j * HID +  0] = acc0[j];
        crow[(size_t)j * HID + 16] = acc1[j];
        crow[(size_t)j * HID + 32] = acc2[j];
        crow[(size_t)j * HID + 48] = acc3[j];
    }
}

// ------------------------------------------------------- edge message pass
// One wave per edge; lane handles 2 features. Gather h[src] (L2-resident,
// 25.6MB) and scatter-add into agg[dst] via f32 atomics (L2 atomic units).
__global__ void edge_msg_k(const int* __restrict__ src, const int* __restrict__ dst,
                           const float* __restrict__ nrm, const float* __restrict__ h,
                           float* __restrict__ agg, int E) {
    int t = blockIdx.x * blockDim.x + threadIdx.x;
    int e = t >> 5;
    if (e >= E) return;
    int lane = t & 31;
    int s = src[e], d = dst[e];
    float nv = nrm[e];
    int f = lane << 1;
    float2 hv = *(const float2*)(h + (size_t)s * HID + f);
    atomicAdd(&agg[(size_t)d * HID + f    ], hv.x * nv);
    atomicAdd(&agg[(size_t)d * HID + f + 1], hv.y * nv);
}

// self-loop (h*dis^2) + bias + ReLU; may write in place over agg.
__global__ void finalize_k(const float* agg, const float* __restrict__ hw,
                           const float* __restrict__ dis, const float* __restrict__ bias,
                           float* outp, int n) {
    int i = blockIdx.x * blockDim.x + threadIdx.x;      // n*16 float4 groups
    if (i >= n * 16) return;
    int node = i >> 4;
    int f4   = (i & 15) << 2;
    float s = dis[node]; s = s * s;
    const float4 a = *(const float4*)(agg + (size_t)node * HID + f4);
    const float4 h = *(const float4*)(hw  + (size_t)node * HID + f4);
    float4 r;
    r.x = fmaxf(a.x + h.x * s + bias[f4 + 0], 0.0f);
    r.y = fmaxf(a.y + h.y * s + bias[f4 + 1], 0.0f);
    r.z = fmaxf(a.z + h.z * s + bias[f4 + 2], 0.0f);
    r.w = fmaxf(a.w + h.w * s + bias[f4 + 3], 0.0f);
    *(float4*)(outp + (size_t)node * HID + f4) = r;
}

// ---------------------------------------------------------------- pooling
__global__ void counts_k(const int* __restrict__ batch, float* __restrict__ counts, int n) {
    int i = blockIdx.x * blockDim.x + threadIdx.x;
    if (i < n) atomicAdd(&counts[batch[i]], 1.0f);
}

// pooled[g][0:64] = sum (-> mean later), pooled[g][64:128] = max (uint-bit trick, h>=0)
__global__ void pool_k(const float* __restrict__ h, const int* __restrict__ batch,
                       float* __restrict__ pooled, int n) {
    int i = blockIdx.x * blockDim.x + threadIdx.x;      // n*64 elements
    if (i >= n * HID) return;
    int node = i >> 6;
    int f    = i & 63;
    int g    = batch[node];
    float v  = h[i];
    atomicAdd(&pooled[(size_t)g * 128 + f], v);
    atomicMax((unsigned int*)&pooled[(size_t)g * 128 + 64 + f], __float_as_uint(v));
}

__global__ void meandiv_k(float* __restrict__ pooled, const float* __restrict__ counts, int G) {
    int i = blockIdx.x * blockDim.x + threadIdx.x;      // G*64
    if (i >= G * HID) return;
    int g = i >> 6;
    int f = i & 63;
    pooled[(size_t)g * 128 + f] /= fmaxf(counts[g], 1.0f);
}

// out[g] = pooled[g,:] . Wout + bout ; one wave32 per graph
__global__ void head_k(const float* __restrict__ pooled, const float* __restrict__ Wout,
                       const float* __restrict__ bout, float* __restrict__ out, int G) {
    int t = blockIdx.x * blockDim.x + threadIdx.x;
    int g = t >> 5;
    int lane = t & 31;
    if (g >= G) return;
    float v = 0.0f;
#pragma unroll
    for (int j = lane; j < 2 * HID; j += 32) v += pooled[(size_t)g * 128 + j] * Wout[j];
#pragma unroll
    for (int off = 16; off; off >>= 1) v += __shfl_xor(v, off, 32);
    if (lane == 0) out[g] = v + bout[0];
}

// ---------------------------------------------------------------- launcher
extern "C" void kernel_launch(void* const* d_in, const int* in_sizes, int n_in,
                              void* d_out, int out_size, void* d_ws, size_t ws_size,
                              hipStream_t stream) {
    const float* x     = (const float*)d_in[0];
    const int*   eidx  = (const int*)  d_in[1];
    const int*   batch = (const int*)  d_in[2];
    const float* Ws[4] = { (const float*)d_in[3], (const float*)d_in[5],
                           (const float*)d_in[7], (const float*)d_in[9] };
    const float* bs[4] = { (const float*)d_in[4], (const float*)d_in[6],
                           (const float*)d_in[8], (const float*)d_in[10] };
    const float* Wout  = (const float*)d_in[11];
    const float* bout  = (const float*)d_in[12];

    const int N = N_NODES, E = N_EDGES, G = NGRAPH;
    const int* src = eidx;
    const int* dst = eidx + E;

    float* ws     = (float*)d_ws;
    float* dis    = ws;                          // N
    float* nrm    = dis + N;                     // E
    float* counts = nrm + E;                     // G
    float* buf0   = counts + G;                  // N*64 (16B aligned: (N+E+G)*4 % 16 == 0)
    float* buf1   = buf0 + (size_t)N * HID;
    float* buf2   = buf1 + (size_t)N * HID;

    float* out    = (float*)d_out;               // [G]
    float* pooled = out + G;                     // [G,128]

    // degree (self-loop => init 1), dis = deg^-1/2, per-edge norm (reused 4x)
    fill_k <<<(N + 255) / 256, 256, 0, stream>>>(dis, 1.0f, N);
    deg_k  <<<(E + 255) / 256, 256, 0, stream>>>(dst, dis, E);
    rsqrt_k<<<(N + 255) / 256, 256, 0, stream>>>(dis, N);
    norm_k <<<(E + 255) / 256, 256, 0, stream>>>(src, dst, dis, nrm, E);

    const float* hin = x;
    float* HW  = buf0;
    float* AGG = buf1;
    float* spare = buf2;
    int K = 128;
    const long long feat = (long long)N * HID;

    for (int l = 0; l < 4; ++l) {
        int gemmThreads = (N / 16) * 32;                     // one wave per 16-row stripe
        gemm_wmma_k<<<(gemmThreads + 255) / 256, 256, 0, stream>>>(hin, Ws[l], HW, N, K);
        fill_k     <<<(int)((feat + 255) / 256), 256, 0, stream>>>(AGG, 0.0f, feat);
        edge_msg_k <<<(int)(((long long)E * 32 + 255) / 256), 256, 0, stream>>>(src, dst, nrm, HW, AGG, E);
        finalize_k <<<(N * 16 + 255) / 256, 256, 0, stream>>>(AGG, HW, dis, bs[l], AGG, N);

        const float* next_in = AGG;
        float* newAGG = (l == 0) ? spare : (float*)hin;      // recycle previous input buffer
        hin = next_in;
        AGG = newAGG;
        K = HID;
    }

    // pooling + head; zero out + pooled (contiguous) and counts first
    fill_k   <<<(G + G * 128 + 255) / 256, 256, 0, stream>>>(out, 0.0f, G + G * 128);
    fill_k   <<<(G + 255) / 256, 256, 0, stream>>>(counts, 0.0f, G);
    counts_k <<<(N + 255) / 256, 256, 0, stream>>>(batch, counts, N);
    pool_k   <<<(N * HID + 255) / 256, 256, 0, stream>>>(hin, batch, pooled, N);
    meandiv_k<<<(G * HID + 255) / 256, 256, 0, stream>>>(pooled, counts, G);
    head_k   <<<(G * 32 + 255) / 256, 256, 0, stream>>>(pooled, Wout, bout, out, G);
}